// VectorQuantizer_6253472383347
// MI455X (gfx1250) — compile-verified
//
#include <hip/hip_runtime.h>

// ---------------------------------------------------------------------------
// VQ-VAE vector quantizer for MI455X (gfx1250, wave32, WMMA + async-to-LDS).
//   z:        (16, 256, 32, 32) f32   -> zf[n][c] = z[b*262144 + c*1024 + hw]
//   codebook: (8192, 256) f32
// Outputs (flat, in return order): z_q (4194304 f32), loss (1 f32),
//                                  idx (16384, written as float)
// ---------------------------------------------------------------------------

#define N_ROWS   16384      // 16*32*32
#define N_EMBED  8192
#define DIM      256
#define KTILES   (N_EMBED / 16)   // 512
#define ZQ_ELEMS 4194304          // 16*256*32*32
#define LOSS_OFF ZQ_ELEMS
#define IDX_OFF  (ZQ_ELEMS + 1)

typedef __attribute__((ext_vector_type(2))) float v2f;
typedef __attribute__((ext_vector_type(8))) float v8f;

// ---------------------------------------------------------------------------
// Kernel 1: codebook squared norms. One wave per row (8 rows / 256-thr block).
// ---------------------------------------------------------------------------
__global__ __launch_bounds__(256) void vq_norms_kernel(
    const float* __restrict__ cb, float* __restrict__ enorm)
{
    const int lane = threadIdx.x & 31;
    const int row  = blockIdx.x * 8 + (threadIdx.x >> 5);
    const float* r = cb + (size_t)row * DIM + lane * 8;
    float4 a = *(const float4*)(r);
    float4 b = *(const float4*)(r + 4);
    float s = a.x*a.x + a.y*a.y + a.z*a.z + a.w*a.w
            + b.x*b.x + b.y*b.y + b.z*b.z + b.w*b.w;
    #pragma unroll
    for (int off = 16; off > 0; off >>= 1) s += __shfl_xor(s, off, 32);
    if (lane == 0) enorm[row] = s;
}

// ---------------------------------------------------------------------------
// Kernel 2: fused distance GEMM (f32 WMMA 16x16x4) + argmin over 8192 codes.
// Block = 256 threads = 8 waves; each wave owns 16 consecutive z rows.
// Grid = 128 blocks (128 rows per block).
//
// Codebook tiles stream global -> LDS via GLOBAL_LOAD_ASYNC_TO_LDS_B128
// (ASYNCcnt), 3-deep buffered: no VGPR staging, full-tile latency cover.
// Async ops complete in order, so "s_wait_asynccnt 4" at the top of iter kt
// guarantees tile kt's 4 ops landed while tile kt+1's remain in flight.
//
// Per WMMA step s (K = 4s..4s+3):
//   A (codebook tile, 16x4): lane l -> row l%16, cols 4s + 2*(l/16) + {0,1}
//                            (contiguous pair -> ds_load_2addr_b64, padded LDS)
//   B (z rows,       4x16): lane l -> col (n) l%16, rows 4s + 2*(l/16) + {0,1}
//                            (register-resident, preloaded once from global)
//   D (16x16 f32): lane l, vgpr r -> (M = r + 8*(l/16) = code, N = l%16 = n)
// ---------------------------------------------------------------------------
__global__ __launch_bounds__(256) void vq_argmin_kernel(
    const float* __restrict__ z, const float* __restrict__ cb,
    const float* __restrict__ enorm,
    int* __restrict__ idx_int, float* __restrict__ idx_f)
{
    __shared__ __align__(16) float cbuf[3][16][260];   // padded rows, 3-deep

    const int tid   = threadIdx.x;
    const int lane  = tid & 31;
    const int wave  = tid >> 5;
    const int lmod  = lane & 15;
    const int lhalf = lane >> 4;

    // This lane's z row (N dimension of the WMMA tile).
    const int n  = blockIdx.x * 128 + wave * 16 + lmod;
    const int bb = n >> 10;
    const int hw = n & 1023;
    const float* zrow = z + (size_t)bb * (DIM * 1024) + hw;  // elem c at zrow[c*1024]

    // ---- Preload all 64 B-fragments (z) into registers: 128 VGPRs, one-time.
    v2f bfrag[64];
    #pragma unroll
    for (int s = 0; s < 64; ++s) {
        const int c0 = 4 * s + 2 * lhalf;
        v2f f;
        f.x = zrow[(size_t)c0 * 1024];
        f.y = zrow[(size_t)(c0 + 1) * 1024];
        bfrag[s] = f;
    }

    // ---- Async codebook tile staging: 16 rows x 256 f32 = 16 KB per tile,
    // 4 x b128 per thread, straight into LDS (IOFFSET applies to both the
    // global source and the LDS destination address).
    const int srow = tid >> 4;   // 0..15 : code row within tile
    const int sseg = tid & 15;   // 0..15 : 16-float segment
    auto stage_async = [&](int kt) {
        const float* src = cb + (size_t)(kt * 16 + srow) * DIM + sseg * 16;
        unsigned dst = (unsigned)(size_t)&cbuf[kt % 3][srow][sseg * 16];
        asm volatile(
            "global_load_async_to_lds_b128 %0, %1, off\n\t"
            "global_load_async_to_lds_b128 %0, %1, off offset:16\n\t"
            "global_load_async_to_lds_b128 %0, %1, off offset:32\n\t"
            "global_load_async_to_lds_b128 %0, %1, off offset:48"
            :: "v"(dst), "v"(src)
            : "memory");
    };

    stage_async(0);
    stage_async(1);

    float minval = 3.402823466e+38f;
    int   minidx = 0;

    for (int kt = 0; kt < KTILES; ++kt) {
        // Tile kt's async ops are the oldest outstanding; in-order completion
        // means ASYNCcnt <= 4 implies they landed (<= 0 on the last tile,
        // where no tile kt+1 is in flight behind them).
        if (kt + 1 < KTILES) {
            asm volatile("s_wait_asynccnt 4" ::: "memory");
        } else {
            asm volatile("s_wait_asynccnt 0" ::: "memory");
        }
        __syncthreads();                       // publish tile kt block-wide

        // Prefetch tile kt+2 into buf[(kt+2)%3]: that buffer was last
        // consumed in iter kt-1, and every thread passed the barrier above
        // after finishing it. Full 64-WMMA tile of latency cover.
        if (kt + 2 < KTILES) stage_async(kt + 2);

        const float* arow = &cbuf[kt % 3][lmod][2 * lhalf];
        v8f acc = {};
        #pragma unroll
        for (int s = 0; s < 64; ++s) {
            v2f a;
            a.x = arow[4 * s];                 // ds_load_2addr_b64 pairs
            a.y = arow[4 * s + 1];
            acc = __builtin_amdgcn_wmma_f32_16x16x4_f32(
                false, a, false, bfrag[s], (short)0, acc, false, false);
        }

        // Fused argmin epilogue: d = ||e_k||^2 - 2 z.e_k  (||z||^2 constant in k)
        const float* en = enorm + kt * 16 + 8 * lhalf;
        const float4 e0 = *(const float4*)(en);
        const float4 e1 = *(const float4*)(en + 4);
        const float ev[8] = { e0.x, e0.y, e0.z, e0.w, e1.x, e1.y, e1.z, e1.w };
        #pragma unroll
        for (int r = 0; r < 8; ++r) {
            const float val = ev[r] - 2.0f * acc[r];
            const int   k   = kt * 16 + 8 * lhalf + r;   // ascending within lane
            if (val < minval) { minval = val; minidx = k; }
        }
    }

    // Combine the two half-wave candidates for this n (lanes l and l+16).
    const float oval = __shfl_xor(minval, 16, 32);
    const int   oidx = __shfl_xor(minidx, 16, 32);
    if (oval < minval || (oval == minval && oidx < minidx)) {
        minval = oval; minidx = oidx;
    }
    if (lhalf == 0) {
        idx_int[n] = minidx;
        idx_f[n]   = (float)minidx;
    }
}

// ---------------------------------------------------------------------------
// Kernel 3: z_q gather with straight-through forward + fused loss partials.
// t indexes output layout (b, c, h, w); writes coalesced, codebook gathered
// from L2. partial[block] = sum over block of (cb - zp)^2.
// ---------------------------------------------------------------------------
__global__ __launch_bounds__(256) void vq_scatter_kernel(
    const float* __restrict__ z, const float* __restrict__ cb,
    const int* __restrict__ idx_int, float* __restrict__ zq,
    float* __restrict__ partial)
{
    const int t  = blockIdx.x * 256 + threadIdx.x;
    const int b  = t >> 18;
    const int c  = (t >> 10) & 255;
    const int hw = t & 1023;
    const int n  = (b << 10) | hw;

    const int   k   = idx_int[n];
    const float zp  = z[t];
    const float cbv = cb[(size_t)k * DIM + c];
    const float d   = cbv - zp;
    zq[t] = zp + d;                 // STE: zp + stop_grad(z_q - zp)

    float s = d * d;
    #pragma unroll
    for (int off = 16; off > 0; off >>= 1) s += __shfl_xor(s, off, 32);
    __shared__ float wsum[8];
    if ((threadIdx.x & 31) == 0) wsum[threadIdx.x >> 5] = s;
    __syncthreads();
    if (threadIdx.x == 0) {
        float tot = 0.f;
        #pragma unroll
        for (int i = 0; i < 8; ++i) tot += wsum[i];
        partial[blockIdx.x] = tot;
    }
}

// ---------------------------------------------------------------------------
// Kernel 4: deterministic final loss reduction (single block, fixed order).
// loss = (BETA + 1) * mean = 1.25/4194304 * sum.
// ---------------------------------------------------------------------------
__global__ __launch_bounds__(256) void vq_loss_kernel(
    const float* __restrict__ partial, float* __restrict__ loss_out)
{
    float s = 0.f;
    for (int i = threadIdx.x; i < 16384; i += 256) s += partial[i];
    #pragma unroll
    for (int off = 16; off > 0; off >>= 1) s += __shfl_xor(s, off, 32);
    __shared__ float wsum[8];
    if ((threadIdx.x & 31) == 0) wsum[threadIdx.x >> 5] = s;
    __syncthreads();
    if (threadIdx.x == 0) {
        float tot = 0.f;
        #pragma unroll
        for (int i = 0; i < 8; ++i) tot += wsum[i];
        *loss_out = tot * (1.25f / 4194304.0f);
    }
}

// ---------------------------------------------------------------------------
extern "C" void kernel_launch(void* const* d_in, const int* in_sizes, int n_in,
                              void* d_out, int out_size, void* d_ws, size_t ws_size,
                              hipStream_t stream)
{
    (void)in_sizes; (void)n_in; (void)out_size; (void)ws_size;

    const float* z  = (const float*)d_in[0];   // (16,256,32,32) f32
    const float* cb = (const float*)d_in[1];   // (8192,256) f32

    float* out   = (float*)d_out;
    float* zq    = out;                        // 4194304
    float* loss  = out + LOSS_OFF;             // 1
    float* idx_f = out + IDX_OFF;              // 16384 (as float)

    float* enorm   = (float*)d_ws;                              // 8192 f32
    int*   idx_int = (int*)((char*)d_ws + N_EMBED * 4);         // 16384 i32
    float* partial = (float*)((char*)d_ws + N_EMBED * 4 + N_ROWS * 4); // 16384 f32

    vq_norms_kernel  <<<N_EMBED / 8, 256, 0, stream>>>(cb, enorm);
    vq_argmin_kernel <<<N_ROWS / 128, 256, 0, stream>>>(z, cb, enorm, idx_int, idx_f);
    vq_scatter_kernel<<<ZQ_ELEMS / 256, 256, 0, stream>>>(z, cb, idx_int, zq, partial);
    vq_loss_kernel   <<<1, 256, 0, stream>>>(partial, loss);
}